// gru_62328565400292
// MI455X (gfx1250) — compile-verified
//
#include <hip/hip_runtime.h>
#include <math.h>

typedef __attribute__((ext_vector_type(2))) float v2f;
typedef __attribute__((ext_vector_type(8))) float v8f;

#define N_INP 512
#define N_HID 512
#define BATCH 64
#define STEPS 512
#define ROWS  (BATCH * STEPS)   // 32768
#define GATE3 (3 * N_HID)       // 1536

#define LDSA_STRIDE 34          // even -> 8B-aligned rows, conflict-free reads
#define LDSB_STRIDE 34

__device__ __forceinline__ v8f wmma_f32(v2f a, v2f b, v8f c) {
    // D = A(16x4,f32) * B(4x16,f32) + C(16x16,f32)
    return __builtin_amdgcn_wmma_f32_16x16x4_f32(
        /*neg_a=*/false, a, /*neg_b=*/false, b,
        /*c_mod=*/(short)0, c, /*reuse_a=*/false, /*reuse_b=*/false);
}

// ---------------- init: zero H and barrier state ----------------
__global__ void init_state(float* Hbuf, unsigned* bar) {
    int i = blockIdx.x * blockDim.x + threadIdx.x;
    if (i < BATCH * N_HID) Hbuf[i] = 0.0f;
    if (i < 2) bar[i] = 0u;
}

// ------------- transpose the three hidden weights (512x512 each) -------------
// WT[w][n][k] = W[w][k][n]  -> B-operand K-pairs become contiguous (b64 loads)
__global__ void transpose3(const float* __restrict__ Whz,
                           const float* __restrict__ Whr,
                           const float* __restrict__ Whh,
                           float* __restrict__ WT) {
    int i = blockIdx.x * 256 + threadIdx.x;      // 0 .. 3*512*512-1
    int w = i >> 18;
    int r = (i >> 9) & (N_HID - 1);
    int c = i & (N_HID - 1);
    const float* W = (w == 0) ? Whz : ((w == 1) ? Whr : Whh);
    WT[i] = W[c * N_HID + r];
}

// ------------- big parallel GEMM: Xproj = X @ [Wxz|Wxr|Wxh] + bias -------------
// block tile 128x64, 8 waves, each wave: 16 rows x 64 cols (4 WMMA tiles)
__global__ __launch_bounds__(256)
void xproj_gemm(const float* __restrict__ X,
                const float* __restrict__ Wxz, const float* __restrict__ Wxr,
                const float* __restrict__ Wxh,
                const float* __restrict__ bz, const float* __restrict__ br,
                const float* __restrict__ bh,
                float* __restrict__ Xproj) {
    __shared__ float As[128 * LDSA_STRIDE];      // [row][k]
    __shared__ float Bs[64 * LDSB_STRIDE];       // [n][k]  (transposed)

    const int tid  = threadIdx.x;
    const int lane = tid & 31;
    const int wv   = tid >> 5;
    const int m    = lane & 15;
    const int hi   = lane >> 4;

    const int rowBase = blockIdx.x * 128;
    const int colBase = blockIdx.y * 64;        // within 1536
    const int gate    = colBase >> 9;           // 0=z 1=r 2=h
    const int gcol    = colBase & (N_HID - 1);
    const float* Wg = (gate == 0) ? Wxz : ((gate == 1) ? Wxr : Wxh);
    const float* bg = (gate == 0) ? bz  : ((gate == 1) ? br  : bh);

    v8f acc0 = {}, acc1 = {}, acc2 = {}, acc3 = {};

    for (int k0 = 0; k0 < N_INP; k0 += 32) {
        // stage A chunk 128x32 as v2f (coalesced, aligned b64 global + LDS)
        #pragma unroll
        for (int i = 0; i < 8; ++i) {
            int e  = tid + i * 256;              // v2f element id, 0..2047
            int r  = e >> 4;                     // 0..127
            int c2 = e & 15;                     // v2f col, 0..15
            v2f v  = *(const v2f*)(X + (size_t)(rowBase + r) * N_INP + k0 + 2 * c2);
            *(v2f*)(&As[r * LDSA_STRIDE + 2 * c2]) = v;
        }
        // stage B chunk 32x64 transposed into Bs[n][k]
        #pragma unroll
        for (int i = 0; i < 4; ++i) {
            int e  = tid + i * 256;              // 0..1023
            int r  = e >> 5;                     // k row, 0..31
            int c2 = e & 31;                     // v2f col pair base
            v2f v  = *(const v2f*)(Wg + (size_t)(k0 + r) * N_HID + gcol + 2 * c2);
            Bs[(2 * c2)     * LDSB_STRIDE + r] = v.x;
            Bs[(2 * c2 + 1) * LDSB_STRIDE + r] = v.y;
        }
        __syncthreads();

        #pragma unroll
        for (int kk = 0; kk < 32; kk += 4) {
            int ka = kk + hi * 2;
            v2f a  = *(const v2f*)(&As[(wv * 16 + m) * LDSA_STRIDE + ka]);
            v2f b0 = *(const v2f*)(&Bs[(m)      * LDSB_STRIDE + ka]);
            v2f b1 = *(const v2f*)(&Bs[(16 + m) * LDSB_STRIDE + ka]);
            v2f b2 = *(const v2f*)(&Bs[(32 + m) * LDSB_STRIDE + ka]);
            v2f b3 = *(const v2f*)(&Bs[(48 + m) * LDSB_STRIDE + ka]);
            acc0 = wmma_f32(a, b0, acc0);
            acc1 = wmma_f32(a, b1, acc1);
            acc2 = wmma_f32(a, b2, acc2);
            acc3 = wmma_f32(a, b3, acc3);
        }
        __syncthreads();
    }

    v8f accs[4] = {acc0, acc1, acc2, acc3};
    #pragma unroll
    for (int tIdx = 0; tIdx < 4; ++tIdx) {
        float bias = bg[gcol + tIdx * 16 + m];
        #pragma unroll
        for (int j = 0; j < 8; ++j) {
            int row = rowBase + wv * 16 + j + 8 * hi;
            int col = colBase + tIdx * 16 + m;
            Xproj[(size_t)row * GATE3 + col] = accs[tIdx][j] + bias;
        }
    }
}

// ---------------- grid-wide barrier (counter + generation) ----------------
__device__ __forceinline__ void grid_barrier(unsigned* bar, unsigned nblocks) {
    __syncthreads();
    if (threadIdx.x == 0) {
        unsigned* cnt = bar;
        unsigned* gen = bar + 1;
        unsigned g = __hip_atomic_load(gen, __ATOMIC_RELAXED, __HIP_MEMORY_SCOPE_AGENT);
        __threadfence();
        unsigned arrived =
            __hip_atomic_fetch_add(cnt, 1u, __ATOMIC_ACQ_REL, __HIP_MEMORY_SCOPE_AGENT);
        if (arrived == nblocks - 1) {
            __hip_atomic_store(cnt, 0u, __ATOMIC_RELAXED, __HIP_MEMORY_SCOPE_AGENT);
            __hip_atomic_fetch_add(gen, 1u, __ATOMIC_RELEASE, __HIP_MEMORY_SCOPE_AGENT);
        } else {
            while (__hip_atomic_load(gen, __ATOMIC_ACQUIRE, __HIP_MEMORY_SCOPE_AGENT) == g) {
                __builtin_amdgcn_s_sleep(1);
            }
        }
        __threadfence();
    }
    __syncthreads();
}

// 512-deep K reduction with TWO independent WMMA accumulator chains for ILP.
__device__ __forceinline__ v8f gemm_tile_k512(const float* __restrict__ arow,
                                              const float* __restrict__ brow,
                                              int hi) {
    v8f accA = {}, accB = {};
    #pragma unroll 4
    for (int k = 0; k < N_HID; k += 8) {
        int ka = k + hi * 2;
        v2f a0 = *(const v2f*)(arow + ka);
        v2f b0 = *(const v2f*)(brow + ka);
        accA = wmma_f32(a0, b0, accA);
        int kb = k + 4 + hi * 2;
        v2f a1 = *(const v2f*)(arow + kb);
        v2f b1 = *(const v2f*)(brow + kb);
        accB = wmma_f32(a1, b1, accB);
    }
    return accA + accB;
}

// ---------------- persistent recurrent kernel ----------------
// 32 blocks x 256 threads = 256 waves.
// Phase A: wave gw owns one 16x16 tile of Z (gw<128) or R (gw>=128).
// Phase B: waves 0..127 own one 16x16 tile of H-tilde / H update.
__global__ __launch_bounds__(256)
void gru_recurrent(const float* __restrict__ Xproj,
                   const float* __restrict__ WhzT, const float* __restrict__ WhrT,
                   const float* __restrict__ WhhT,
                   float* __restrict__ Hbuf, float* __restrict__ Zbuf,
                   float* __restrict__ RHbuf,
                   float* __restrict__ out, float* __restrict__ hfinal,
                   unsigned* bar) {
    const int tid  = threadIdx.x;
    const int lane = tid & 31;
    const int wv   = tid >> 5;
    const int gw   = blockIdx.x * 8 + wv;   // 0..255
    const int m    = lane & 15;
    const int hi   = lane >> 4;
    const unsigned nblk = gridDim.x;

    for (int t = 0; t < STEPS; ++t) {
        // ---- Phase A: Z and R gates (wave-uniform control; EXEC stays full) ----
        {
            const int gate = gw >> 7;          // 0 = z, 1 = r
            const int tile = gw & 127;
            const int row0 = (tile >> 5) * 16;
            const int col0 = (tile & 31) * 16;
            const float* WT = gate ? WhrT : WhzT;
            v8f acc = gemm_tile_k512(Hbuf + (row0 + m) * N_HID,
                                     WT   + (col0 + m) * N_HID, hi);
            #pragma unroll
            for (int j = 0; j < 8; ++j) {
                int i = row0 + j + 8 * hi;     // batch row
                int c = col0 + m;              // hidden col
                float pre = acc[j] +
                    Xproj[((size_t)i * STEPS + t) * GATE3 + gate * N_HID + c];
                float s = 1.0f / (1.0f + __expf(-pre));
                if (gate == 0) Zbuf[i * N_HID + c] = s;
                else           RHbuf[i * N_HID + c] = s * Hbuf[i * N_HID + c];
            }
        }
        grid_barrier(bar, nblk);

        // ---- Phase B: H_tilde = tanh((R*H)@Whh + xh), blend, write output ----
        if (gw < 128) {
            const int row0 = (gw >> 5) * 16;
            const int col0 = (gw & 31) * 16;
            v8f acc = gemm_tile_k512(RHbuf + (row0 + m) * N_HID,
                                     WhhT  + (col0 + m) * N_HID, hi);
            #pragma unroll
            for (int j = 0; j < 8; ++j) {
                int i = row0 + j + 8 * hi;
                int c = col0 + m;
                float pre = acc[j] +
                    Xproj[((size_t)i * STEPS + t) * GATE3 + 2 * N_HID + c];
                float ht   = tanhf(pre);
                float z    = Zbuf[i * N_HID + c];
                float hold = Hbuf[i * N_HID + c];
                float hn   = z * hold + (1.0f - z) * ht;
                Hbuf[i * N_HID + c] = hn;
                out[((size_t)i * STEPS + t) * N_HID + c] = hn;
                if (t == STEPS - 1) hfinal[i * N_HID + c] = hn;
            }
        }
        grid_barrier(bar, nblk);
    }
}

extern "C" void kernel_launch(void* const* d_in, const int* in_sizes, int n_in,
                              void* d_out, int out_size, void* d_ws, size_t ws_size,
                              hipStream_t stream) {
    const float* X   = (const float*)d_in[0];
    const float* Wxz = (const float*)d_in[1];
    const float* Whz = (const float*)d_in[2];
    const float* bz  = (const float*)d_in[3];
    const float* Wxr = (const float*)d_in[4];
    const float* Whr = (const float*)d_in[5];
    const float* br  = (const float*)d_in[6];
    const float* Wxh = (const float*)d_in[7];
    const float* Whh = (const float*)d_in[8];
    const float* bh  = (const float*)d_in[9];

    float* ws     = (float*)d_ws;
    float* Xproj  = ws;                                    // 32768*1536 floats
    float* WhzT   = Xproj + (size_t)ROWS * GATE3;
    float* WhrT   = WhzT + (size_t)N_HID * N_HID;
    float* WhhT   = WhrT + (size_t)N_HID * N_HID;
    float* Hbuf   = WhhT + (size_t)N_HID * N_HID;
    float* Zbuf   = Hbuf + BATCH * N_HID;
    float* RHbuf  = Zbuf + BATCH * N_HID;
    unsigned* bar = (unsigned*)(RHbuf + BATCH * N_HID);

    float* out    = (float*)d_out;
    float* hfinal = out + (size_t)BATCH * STEPS * N_HID;

    init_state<<<(BATCH * N_HID + 255) / 256, 256, 0, stream>>>(Hbuf, bar);
    transpose3<<<3 * N_HID * N_HID / 256, 256, 0, stream>>>(Whz, Whr, Whh, WhzT);
    dim3 g1(ROWS / 128, GATE3 / 64);
    xproj_gemm<<<g1, 256, 0, stream>>>(X, Wxz, Wxr, Wxh, bz, br, bh, Xproj);
    gru_recurrent<<<32, 256, 0, stream>>>(Xproj, WhzT, WhrT, WhhT,
                                          Hbuf, Zbuf, RHbuf, out, hfinal, bar);
}